// Model_72696616452481
// MI455X (gfx1250) — compile-verified
//
#include <hip/hip_runtime.h>

#define N_NODES 50000
#define IN_C 300
#define HID 64
#define OUT_C 40
#define N_EDGES 800000
#define ALPHA 1.0f
#define MAX_ITER 15
#define NEUMANN_K 10
#define GAMMA 0.8f

#define N_TILES (N_NODES / 16)   // 3125 exactly
#define WAVES_PER_BLOCK 8
#define W_STRIDE 66              // LDS row pad: banks = (66n+k)%64 -> conflict-free

typedef __attribute__((ext_vector_type(2))) float v2f;
typedef __attribute__((ext_vector_type(8))) float v8f;

__device__ __forceinline__ v8f wmma_f32_k4(v2f a, v2f b, v8f c) {
    // V_WMMA_F32_16X16X4_F32 : D = A(16x4) * B(4x16) + C(16x16), fp32 throughout
    return __builtin_amdgcn_wmma_f32_16x16x4_f32(
        /*neg_a=*/false, a, /*neg_b=*/false, b,
        /*c_mod=*/(short)0, c, /*reuse_a=*/false, /*reuse_b=*/false);
}

// ---------------------------------------------------------------------------
// W = GAMMA * (I+S)^{-1} (I-S),  S = 0.5*(M - M^T).  One block, 64 threads.
// I+S with S skew-symmetric has det >= 1 -> no pivoting required.
// ---------------------------------------------------------------------------
__global__ void cayley_kernel(const float* __restrict__ M, float* __restrict__ Wout) {
    __shared__ float A[64][64];
    __shared__ float B[64][64];
    __shared__ float fcol[64];
    const int tid = threadIdx.x;  // 0..63, owns column tid
    for (int i = 0; i < 64; ++i) {
        float s = 0.5f * (M[i * 64 + tid] - M[tid * 64 + i]);
        float d = (i == tid) ? 1.0f : 0.0f;
        A[i][tid] = d + s;
        B[i][tid] = d - s;
    }
    __syncthreads();
    for (int p = 0; p < 64; ++p) {
        float piv = A[p][p];
        __syncthreads();
        float inv = 1.0f / piv;
        A[p][tid] *= inv;
        B[p][tid] *= inv;
        __syncthreads();
        fcol[tid] = A[tid][p];
        __syncthreads();
        const float ap = A[p][tid];
        const float bp = B[p][tid];
        for (int i = 0; i < 64; ++i) {
            if (i != p) {
                A[i][tid] -= fcol[i] * ap;
                B[i][tid] -= fcol[i] * bp;
            }
        }
        __syncthreads();
    }
    for (int i = 0; i < 64; ++i)
        Wout[i * 64 + tid] = GAMMA * B[i][tid];
}

// ---------------------------------------------------------------------------
// CSR build: deg count -> exclusive scan (1 block) -> scatter into CSR order.
// One-time cost per call (~1.6M atomics) replacing 150 x 51.2M atomics.
// ---------------------------------------------------------------------------
__global__ void count_kernel(const int* __restrict__ row, int* __restrict__ deg) {
    const int e = blockIdx.x * blockDim.x + threadIdx.x;
    if (e < N_EDGES) atomicAdd(&deg[row[e]], 1);
}

__global__ void scan_kernel(const int* __restrict__ deg,
                            int* __restrict__ rowptr, int* __restrict__ cursor) {
    __shared__ int partial[256];
    const int tid = threadIdx.x;                     // 256 threads, 1 block
    const int CHUNK = (N_NODES + 255) / 256;         // 196
    const int base = tid * CHUNK;
    int s = 0;
    for (int i = 0; i < CHUNK; ++i) {
        int idx = base + i;
        if (idx < N_NODES) s += deg[idx];
    }
    partial[tid] = s;
    __syncthreads();
    for (int off = 1; off < 256; off <<= 1) {        // Hillis-Steele inclusive
        int v = (tid >= off) ? partial[tid - off] : 0;
        __syncthreads();
        partial[tid] += v;
        __syncthreads();
    }
    int run = (tid == 0) ? 0 : partial[tid - 1];     // exclusive chunk offset
    for (int i = 0; i < CHUNK; ++i) {
        int idx = base + i;
        if (idx < N_NODES) {
            rowptr[idx] = run;
            cursor[idx] = run;
            run += deg[idx];
        }
    }
    if (tid == 255) rowptr[N_NODES] = run;           // == N_EDGES
}

__global__ void scatter_kernel(const int* __restrict__ row, const int* __restrict__ col,
                               const float* __restrict__ ew, int* __restrict__ cursor,
                               int* __restrict__ col_s, float* __restrict__ w_s) {
    const int e = blockIdx.x * blockDim.x + threadIdx.x;
    if (e >= N_EDGES) return;
    const int p = atomicAdd(&cursor[row[e]], 1);
    col_s[p] = col[e];
    w_s[p]  = ew[e];
}

// ---------------------------------------------------------------------------
// bias = x @ enc_w^T + enc_b + lin_b      (50000 x 300) @ (300 x 64)
// One wave -> one 16x64 strip. WMMA f32 16x16x4, K stepped by 4.
// ---------------------------------------------------------------------------
__global__ void encoder_kernel(const float* __restrict__ x,
                               const float* __restrict__ enc_w,
                               const float* __restrict__ enc_b,
                               const float* __restrict__ lin_b,
                               float* __restrict__ bias_out) {
    const int wave = threadIdx.x >> 5;
    const int lane = threadIdx.x & 31;
    const int tile = blockIdx.x * WAVES_PER_BLOCK + wave;
    if (tile >= N_TILES) return;              // wave-uniform: EXEC stays all-1s
    const int row0 = tile * 16;
    const int half = lane >> 4;               // 0: K0/K1, 1: K2/K3
    const int l    = lane & 15;

    const float* __restrict__ arow = x + (size_t)(row0 + l) * IN_C + 2 * half;
    const float* __restrict__ b0p  = enc_w + (size_t)(l +  0) * IN_C + 2 * half;
    const float* __restrict__ b1p  = enc_w + (size_t)(l + 16) * IN_C + 2 * half;
    const float* __restrict__ b2p  = enc_w + (size_t)(l + 32) * IN_C + 2 * half;
    const float* __restrict__ b3p  = enc_w + (size_t)(l + 48) * IN_C + 2 * half;

    v8f c0 = {}, c1 = {}, c2 = {}, c3 = {};
#pragma unroll 5
    for (int k = 0; k < IN_C; k += 4) {
        v2f a, b0, b1, b2, b3;
        a.x  = arow[k]; a.y  = arow[k + 1];
        b0.x = b0p[k];  b0.y = b0p[k + 1];
        b1.x = b1p[k];  b1.y = b1p[k + 1];
        b2.x = b2p[k];  b2.y = b2p[k + 1];
        b3.x = b3p[k];  b3.y = b3p[k + 1];
        c0 = wmma_f32_k4(a, b0, c0);
        c1 = wmma_f32_k4(a, b1, c1);
        c2 = wmma_f32_k4(a, b2, c2);
        c3 = wmma_f32_k4(a, b3, c3);
    }
    const float e0 = enc_b[l +  0] + lin_b[l +  0];
    const float e1 = enc_b[l + 16] + lin_b[l + 16];
    const float e2 = enc_b[l + 32] + lin_b[l + 32];
    const float e3 = enc_b[l + 48] + lin_b[l + 48];
#pragma unroll
    for (int v = 0; v < 8; ++v) {
        float* orow = bias_out + (size_t)(row0 + v + 8 * half) * HID;
        orow[l +  0] = c0[v] + e0;
        orow[l + 16] = c1[v] + e1;
        orow[l + 32] = c2[v] + e2;
        orow[l + 48] = c3[v] + e3;
    }
}

// ---------------------------------------------------------------------------
// Fused Neumann term:  cur_out = beta * (A_csr @ cur_in) @ W^T;  acc += cur_out
// One wave per 16-node tile. Lane(half,l) accumulates row (row0+l) features
// with k%4 in {2*half, 2*half+1} -- i.e. directly in the WMMA A-fragment
// layout -- then 16 k-steps of V_WMMA_F32_16X16X4_F32 with B read from LDS.
// ---------------------------------------------------------------------------
__global__ void neumann_fused_kernel(const int* __restrict__ rowptr,
                                     const int* __restrict__ col_s,
                                     const float* __restrict__ w_s,
                                     const float* __restrict__ cur_in,
                                     const float* __restrict__ Wm,
                                     float* __restrict__ cur_out,
                                     float* __restrict__ acc,
                                     float beta_scale) {
    __shared__ float Wsh[64 * W_STRIDE];
    // cooperative W load: Wsh[n*W_STRIDE + k] = Wm[n*64 + k]
    for (int idx = threadIdx.x; idx < 64 * 64; idx += blockDim.x) {
        const int n = idx >> 6, k = idx & 63;
        Wsh[n * W_STRIDE + k] = Wm[idx];
    }
    __syncthreads();

    const int wave = threadIdx.x >> 5;
    const int lane = threadIdx.x & 31;
    const int tile = blockIdx.x * WAVES_PER_BLOCK + wave;
    if (tile >= N_TILES) return;              // wave-uniform; no syncthreads below
    const int row0 = tile * 16;
    const int half = lane >> 4;
    const int l    = lane & 15;

    // -------- gather/aggregate this lane's WMMA-A fragment ----------------
    const int r     = row0 + l;
    const int start = rowptr[r];
    const int end   = rowptr[r + 1];
    float2 av[16];
#pragma unroll
    for (int kk = 0; kk < 16; ++kk) { av[kk].x = 0.0f; av[kk].y = 0.0f; }

    for (int e = start; e < end; ++e) {
        const int   c = col_s[e];
        const float w = w_s[e];
        const float* __restrict__ src = cur_in + (size_t)c * HID + 2 * half;
#pragma unroll
        for (int kk = 0; kk < 16; ++kk) {
            const float2 v = *(const float2*)(src + 4 * kk);
            av[kk].x += w * v.x;
            av[kk].y += w * v.y;
        }
    }
    // loop divergence is structured: EXEC is all-1s again here (WMMA-legal)

    // -------- (16x64 agg tile) @ W^T via 16 WMMA k-steps ------------------
    const float* __restrict__ b0p = &Wsh[(l +  0) * W_STRIDE + 2 * half];
    const float* __restrict__ b1p = &Wsh[(l + 16) * W_STRIDE + 2 * half];
    const float* __restrict__ b2p = &Wsh[(l + 32) * W_STRIDE + 2 * half];
    const float* __restrict__ b3p = &Wsh[(l + 48) * W_STRIDE + 2 * half];

    v8f c0 = {}, c1 = {}, c2 = {}, c3 = {};
#pragma unroll
    for (int kk = 0; kk < 16; ++kk) {
        const int kb = 4 * kk;
        v2f a, b0, b1, b2, b3;
        a.x  = av[kk].x;  a.y  = av[kk].y;
        b0.x = b0p[kb];   b0.y = b0p[kb + 1];
        b1.x = b1p[kb];   b1.y = b1p[kb + 1];
        b2.x = b2p[kb];   b2.y = b2p[kb + 1];
        b3.x = b3p[kb];   b3.y = b3p[kb + 1];
        c0 = wmma_f32_k4(a, b0, c0);
        c1 = wmma_f32_k4(a, b1, c1);
        c2 = wmma_f32_k4(a, b2, c2);
        c3 = wmma_f32_k4(a, b3, c3);
    }
#pragma unroll
    for (int v = 0; v < 8; ++v) {
        const size_t rr = (size_t)(row0 + v + 8 * half) * HID;
        const float v0 = beta_scale * c0[v];
        const float v1 = beta_scale * c1[v];
        const float v2 = beta_scale * c2[v];
        const float v3 = beta_scale * c3[v];
        cur_out[rr + l +  0] = v0;  acc[rr + l +  0] += v0;
        cur_out[rr + l + 16] = v1;  acc[rr + l + 16] += v1;
        cur_out[rr + l + 32] = v2;  acc[rr + l + 32] += v2;
        cur_out[rr + l + 48] = v3;  acc[rr + l + 48] += v3;
    }
}

// ---------------------------------------------------------------------------
// out = relu(z) @ dec_w^T   (n x 64) @ (64 x 40). 3 N-tiles, n<40 guarded.
// ---------------------------------------------------------------------------
__global__ void decoder_kernel(const float* __restrict__ z,
                               const float* __restrict__ dec_w,
                               float* __restrict__ out) {
    const int wave = threadIdx.x >> 5;
    const int lane = threadIdx.x & 31;
    const int tile = blockIdx.x * WAVES_PER_BLOCK + wave;
    if (tile >= N_TILES) return;
    const int row0 = tile * 16;
    const int half = lane >> 4;
    const int l    = lane & 15;

    const float* __restrict__ arow = z + (size_t)(row0 + l) * HID + 2 * half;
    const int n0 = l, n1 = l + 16, n2 = l + 32;
    const float* __restrict__ b0p = dec_w + (size_t)n0 * HID + 2 * half;
    const float* __restrict__ b1p = dec_w + (size_t)n1 * HID + 2 * half;
    const float* __restrict__ b2p = (n2 < OUT_C) ? dec_w + (size_t)n2 * HID + 2 * half
                                                 : dec_w + 2 * half;  // dummy
    const bool n2ok = (n2 < OUT_C);

    v8f c0 = {}, c1 = {}, c2 = {};
#pragma unroll
    for (int k = 0; k < HID; k += 4) {
        v2f a, b0, b1, b2;
        a.x  = fmaxf(arow[k], 0.0f);
        a.y  = fmaxf(arow[k + 1], 0.0f);
        b0.x = b0p[k]; b0.y = b0p[k + 1];
        b1.x = b1p[k]; b1.y = b1p[k + 1];
        b2.x = n2ok ? b2p[k] : 0.0f;
        b2.y = n2ok ? b2p[k + 1] : 0.0f;
        c0 = wmma_f32_k4(a, b0, c0);
        c1 = wmma_f32_k4(a, b1, c1);
        c2 = wmma_f32_k4(a, b2, c2);
    }
#pragma unroll
    for (int v = 0; v < 8; ++v) {
        float* orow = out + (size_t)(row0 + v + 8 * half) * OUT_C;
        orow[n0] = c0[v];
        orow[n1] = c1[v];
        if (n2ok) orow[n2] = c2[v];
    }
}

// ---------------------------------------------------------------------------
// Elementwise pieces of the Peaceman-Rachford step
// ---------------------------------------------------------------------------
__global__ void prep_kernel(const float* __restrict__ z,
                            float* __restrict__ u,          // in: u, out: u_half
                            const float* __restrict__ bias,
                            float* __restrict__ cur,
                            float* __restrict__ acc) {
    const size_t i = (size_t)blockIdx.x * blockDim.x + threadIdx.x;
    if (i >= (size_t)N_NODES * HID) return;
    const float u_half = 2.0f * z[i] - u[i];
    u[i] = u_half;
    const float y = u_half + ALPHA * bias[i];
    cur[i] = y;
    acc[i] = y;
}

__global__ void finish_kernel(const float* __restrict__ acc,
                              float* __restrict__ u,        // in: u_half, out: u_new
                              float* __restrict__ z) {
    const size_t i = (size_t)blockIdx.x * blockDim.x + threadIdx.x;
    if (i >= (size_t)N_NODES * HID) return;
    const float z_half = acc[i] * (1.0f / (1.0f + ALPHA));
    const float u_new = 2.0f * z_half - u[i];
    u[i] = u_new;
    z[i] = fmaxf(u_new, 0.0f);
}

// ---------------------------------------------------------------------------
extern "C" void kernel_launch(void* const* d_in, const int* in_sizes, int n_in,
                              void* d_out, int out_size, void* d_ws, size_t ws_size,
                              hipStream_t stream) {
    (void)in_sizes; (void)n_in; (void)out_size; (void)ws_size;
    const float* x      = (const float*)d_in[0];
    const int*   eidx   = (const int*)  d_in[1];
    const float* ew     = (const float*)d_in[2];
    const float* enc_w  = (const float*)d_in[3];
    const float* enc_b  = (const float*)d_in[4];
    const float* dec_w  = (const float*)d_in[5];
    const float* M      = (const float*)d_in[6];
    const float* lin_b  = (const float*)d_in[7];
    float* out = (float*)d_out;

    const size_t NH = (size_t)N_NODES * HID;
    float* Wm   = (float*)d_ws;
    float* bias = Wm + 4096;
    float* z    = bias + NH;
    float* u    = z + NH;
    float* curA = u + NH;
    float* curB = curA + NH;
    float* acc  = curB + NH;
    float* w_s  = acc + NH;                         // N_EDGES floats
    int*   deg     = (int*)(w_s + N_EDGES);         // N_NODES (doubles as cursor)
    int*   rowptr  = deg + N_NODES;                 // N_NODES + 1
    int*   col_s   = rowptr + (N_NODES + 2);        // N_EDGES

    const int* row = eidx;
    const int* col = eidx + N_EDGES;

    const int gemm_grid = (N_TILES + WAVES_PER_BLOCK - 1) / WAVES_PER_BLOCK;
    const int ew_grid   = (int)((NH + 255) / 256);
    const int edge_grid = (N_EDGES + 255) / 256;

    // ---- one-time per call: W, CSR, encoder, state init -------------------
    cayley_kernel<<<1, 64, 0, stream>>>(M, Wm);
    hipMemsetAsync(deg, 0, N_NODES * sizeof(int), stream);
    count_kernel<<<edge_grid, 256, 0, stream>>>(row, deg);
    scan_kernel<<<1, 256, 0, stream>>>(deg, rowptr, deg /*cursor (deg reused)*/);
    scatter_kernel<<<edge_grid, 256, 0, stream>>>(row, col, ew, deg, col_s, w_s);
    encoder_kernel<<<gemm_grid, 256, 0, stream>>>(x, enc_w, enc_b, lin_b, bias);
    hipMemsetAsync(z, 0, NH * sizeof(float), stream);
    hipMemsetAsync(u, 0, NH * sizeof(float), stream);

    // ---- Peaceman-Rachford outer loop -------------------------------------
    const float beta = ALPHA / (1.0f + ALPHA);
    for (int it = 0; it < MAX_ITER; ++it) {
        prep_kernel<<<ew_grid, 256, 0, stream>>>(z, u, bias, curA, acc);
        float* cin = curA;
        float* cout = curB;
        for (int k = 0; k < NEUMANN_K; ++k) {
            neumann_fused_kernel<<<gemm_grid, 256, 0, stream>>>(
                rowptr, col_s, w_s, cin, Wm, cout, acc, beta);
            float* t = cin; cin = cout; cout = t;
        }
        finish_kernel<<<ew_grid, 256, 0, stream>>>(acc, u, z);
    }
    decoder_kernel<<<gemm_grid, 256, 0, stream>>>(z, dec_w, out);
}